// contrast_PIRL_35218731827210
// MI455X (gfx1250) — compile-verified
//
#include <hip/hip_runtime.h>
#include <hip/hip_bf16.h>
#include <math.h>

typedef __attribute__((ext_vector_type(2))) float v2f;
typedef __attribute__((ext_vector_type(8))) float v8f;

#define PIRL_BSZ   64
#define PIRL_NDIM  128
#define PIRL_KTOT  16385          // K+1
#define PIRL_NTILE 1025           // ceil(16385/16)
#define PIRL_NDATA 1000000
#define PIRL_INVT  (1.0f / 0.07f)

// ---------------------------------------------------------------------------
// Kernel A: batched gemv over gathered memory rows via V_WMMA_F32_16X16X4_F32.
// One wave handles one (b, 16-k tile). A = 16 gathered rows (16x4 f32 slices),
// B columns 0..7 = x[b], columns 8..15 = x_jig[b]. After accumulating K=128,
// column 0 of D holds logits, column 8 holds logits_jig (pre 1/T scale).
// ---------------------------------------------------------------------------
__global__ __launch_bounds__(256)
void pirl_logits_wmma(const float* __restrict__ x,
                      const float* __restrict__ x_jig,
                      const int*   __restrict__ y,
                      const int*   __restrict__ idx,
                      const float* __restrict__ memory,
                      float* __restrict__ logits,
                      float* __restrict__ logits_jig)
{
    const int lane      = threadIdx.x & 31;
    const int waveInBlk = threadIdx.x >> 5;
    const long waveId   = (long)blockIdx.x * 8 + waveInBlk;
    const int b = (int)(waveId / PIRL_NTILE);
    const int t = (int)(waveId % PIRL_NTILE);
    if (b >= PIRL_BSZ) return;   // wave-uniform; EXEC stays all-ones for WMMA

    const int n  = lane & 15;    // A: row M; B/D: column N
    const int hi = lane >> 4;

    // A-fragment row for this lane (tail rows clamp to k=16384; results unstored)
    int kk = t * 16 + n;
    int kEff = kk > (PIRL_KTOT - 1) ? (PIRL_KTOT - 1) : kk;
    int gidx = (kEff == 0) ? y[b] : idx[(size_t)b * PIRL_KTOT + kEff];
    const float* arow = memory + (size_t)gidx * PIRL_NDIM + 2 * hi;

    // B-fragment source: columns 0..7 -> x[b], columns 8..15 -> x_jig[b]
    const float* bsrc = ((n < 8) ? x : x_jig) + (size_t)b * PIRL_NDIM + 2 * hi;

    v8f c = {};
    #pragma unroll 4
    for (int d = 0; d < PIRL_NDIM; d += 4) {
        v2f a  = *(const v2f*)(arow + d);   // lane holds A[n][d+2*hi], A[n][d+2*hi+1]
        v2f bb = *(const v2f*)(bsrc + d);   // lane holds B[d+2*hi][n], B[d+2*hi+1][n]
        c = __builtin_amdgcn_wmma_f32_16x16x4_f32(
                /*neg_a=*/false, a, /*neg_b=*/false, bb,
                /*c_mod=*/(short)0, c, /*reuse_a=*/false, /*reuse_b=*/false);
    }

    // D layout: lanes 0-15 = column N, VGPR r = row M=r (lanes 0-15) / M=r+8 (16-31)
    if (n == 0 || n == 8) {
        float* dst = (n == 0) ? logits : logits_jig;
        const size_t base = (size_t)b * PIRL_KTOT;
        const int kbase = t * 16 + hi * 8;
        #pragma unroll
        for (int r = 0; r < 8; ++r) {
            int k2 = kbase + r;
            if (k2 < PIRL_KTOT) dst[base + k2] = c[r] * PIRL_INVT;
        }
    }
}

// ---------------------------------------------------------------------------
// Kernel B: per-(b, which) cross-entropy with target class 0.
// grid = 128 blocks: block id = which*64 + b.
// ---------------------------------------------------------------------------
__global__ __launch_bounds__(256)
void pirl_ce(const float* __restrict__ lg, float* __restrict__ ce)
{
    __shared__ float red[256];
    const int id = blockIdx.x;
    const int which = id >> 6;
    const int b = id & 63;
    const float* src = lg + (size_t)which * ((size_t)PIRL_BSZ * PIRL_KTOT)
                          + (size_t)b * PIRL_KTOT;

    float mx = -3.402823466e38f;
    for (int i = threadIdx.x; i < PIRL_KTOT; i += 256) mx = fmaxf(mx, src[i]);
    red[threadIdx.x] = mx; __syncthreads();
    for (int s = 128; s > 0; s >>= 1) {
        if ((int)threadIdx.x < s) red[threadIdx.x] = fmaxf(red[threadIdx.x], red[threadIdx.x + s]);
        __syncthreads();
    }
    mx = red[0]; __syncthreads();

    float sum = 0.0f;
    for (int i = threadIdx.x; i < PIRL_KTOT; i += 256) sum += expf(src[i] - mx);
    red[threadIdx.x] = sum; __syncthreads();
    for (int s = 128; s > 0; s >>= 1) {
        if ((int)threadIdx.x < s) red[threadIdx.x] += red[threadIdx.x + s];
        __syncthreads();
    }
    if (threadIdx.x == 0) ce[id] = -(src[0] - mx - logf(red[0]));
}

// ---------------------------------------------------------------------------
// Kernel C: loss = (sum of 128 CE values) / 128
//   ( = 0.5*mean_b(ce) + 0.5*mean_b(ce_jig) )
// ---------------------------------------------------------------------------
__global__ void pirl_loss(const float* __restrict__ ce, float* __restrict__ out)
{
    __shared__ float red[128];
    red[threadIdx.x] = ce[threadIdx.x];
    __syncthreads();
    for (int s = 64; s > 0; s >>= 1) {
        if ((int)threadIdx.x < s) red[threadIdx.x] += red[threadIdx.x + s];
        __syncthreads();
    }
    if (threadIdx.x == 0) out[0] = red[0] * (1.0f / 128.0f);
}

// ---------------------------------------------------------------------------
// Kernel D: stream memory bank -> output (b128 copies, grid-stride)
// ---------------------------------------------------------------------------
__global__ __launch_bounds__(256)
void pirl_copy(const float4* __restrict__ src, float4* __restrict__ dst, long n4)
{
    long i = (long)blockIdx.x * blockDim.x + threadIdx.x;
    const long stride = (long)gridDim.x * blockDim.x;
    for (; i < n4; i += stride) dst[i] = src[i];
}

// ---------------------------------------------------------------------------
// Kernel E: overwrite rows y[b] with l2norm(M*memory[y[b]] + (1-M)*x[b]).
// One wave32 per row; lane holds 4 dims; shfl_xor reduction for the norm.
// ---------------------------------------------------------------------------
__global__ __launch_bounds__(256)
void pirl_update(const float* __restrict__ x,
                 const int*   __restrict__ y,
                 const float* __restrict__ memory,
                 float* __restrict__ out_mem)
{
    const int wave = (int)((blockIdx.x * blockDim.x + threadIdx.x) >> 5);
    const int lane = threadIdx.x & 31;
    if (wave >= PIRL_BSZ) return;

    const int row = y[wave];
    const float4 mv = *(const float4*)(memory + (size_t)row  * PIRL_NDIM + lane * 4);
    const float4 xv = *(const float4*)(x      + (size_t)wave * PIRL_NDIM + lane * 4);

    const float MM = 0.5f;  // EMA momentum
    float4 w;
    w.x = mv.x * MM + xv.x * (1.0f - MM);
    w.y = mv.y * MM + xv.y * (1.0f - MM);
    w.z = mv.z * MM + xv.z * (1.0f - MM);
    w.w = mv.w * MM + xv.w * (1.0f - MM);

    float s = w.x * w.x + w.y * w.y + w.z * w.z + w.w * w.w;
    #pragma unroll
    for (int off = 16; off > 0; off >>= 1) s += __shfl_xor(s, off, 32);

    const float inv = 1.0f / fmaxf(sqrtf(s), 1e-12f);
    float4 o = {w.x * inv, w.y * inv, w.z * inv, w.w * inv};
    *(float4*)(out_mem + (size_t)row * PIRL_NDIM + lane * 4) = o;
}

// ---------------------------------------------------------------------------
// Launch: inputs in setup_inputs() order: x, y, x_jig, idx, memory.
// d_out: [loss (1 float)] ++ [new_memory (1e6*128 floats)].
// ---------------------------------------------------------------------------
extern "C" void kernel_launch(void* const* d_in, const int* in_sizes, int n_in,
                              void* d_out, int out_size, void* d_ws, size_t ws_size,
                              hipStream_t stream)
{
    const float* x      = (const float*)d_in[0];
    const int*   y      = (const int*)  d_in[1];
    const float* x_jig  = (const float*)d_in[2];
    const int*   idx    = (const int*)  d_in[3];
    const float* memory = (const float*)d_in[4];

    float* out     = (float*)d_out;
    float* out_mem = out + 1;

    float* ws_logits     = (float*)d_ws;
    float* ws_logits_jig = ws_logits + (size_t)PIRL_BSZ * PIRL_KTOT;
    float* ws_ce         = ws_logits + 2 * (size_t)PIRL_BSZ * PIRL_KTOT;

    // 64 * 1025 waves, 8 waves per 256-thread block -> 8200 blocks exactly
    pirl_logits_wmma<<<8200, 256, 0, stream>>>(x, x_jig, y, idx, memory,
                                               ws_logits, ws_logits_jig);
    pirl_ce<<<128, 256, 0, stream>>>(ws_logits, ws_ce);
    pirl_loss<<<1, 128, 0, stream>>>(ws_ce, out);

    const long n4 = (long)PIRL_NDATA * PIRL_NDIM / 4;   // 32M float4
    pirl_copy<<<8192, 256, 0, stream>>>((const float4*)memory, (float4*)out_mem, n4);
    pirl_update<<<8, 256, 0, stream>>>(x, y, memory, out_mem);
}